// Pointer_31035433681578
// MI455X (gfx1250) — compile-verified
//
#include <hip/hip_runtime.h>
#include <math.h>

// Problem constants (from the reference): B=64, LQ=1, LK=8192, H=128, V=50257
#define BATCH   64
#define LK      8192
#define HDIM    128
#define VOCAB   50257
#define NEG_INF -1000000000.0f

typedef __attribute__((ext_vector_type(2))) float v2f;
typedef __attribute__((ext_vector_type(8))) float v8f;

// ---------------------------------------------------------------------------
// K0: zero the dist region of d_out (harness poisons it with 0xAA each run)
// ---------------------------------------------------------------------------
__global__ void zero_kernel(float* __restrict__ p, size_t n) {
    size_t i = (size_t)blockIdx.x * blockDim.x + threadIdx.x;
    if (i < n) p[i] = 0.0f;
}

// ---------------------------------------------------------------------------
// K1: C[64 x 128] = A[64 x 128] @ B, with B either w (row-major, k*H+n) or
//     w^T (element (k,n) = w[n*H+k]).  One wave (32 threads) per 16x16 tile,
//     accumulated with V_WMMA_F32_16X16X4_F32 (full fp32 precision).
//
// WMMA f32 16x16x4 VGPR layouts (CDNA5 ISA 7.12.2):
//   A (16x4 MxK): lane l<16 -> M=l; VGPR0 = K=(0|2), VGPR1 = K=(1|3) by half
//   B (4x16 KxN): lane l<16 -> N=l; VGPR0 = K=(0|2), VGPR1 = K=(1|3) by half
//   C (16x16):    VGPR g -> M = g + 8*half, N = lane%16
// ---------------------------------------------------------------------------
template <bool TRANSB>
__global__ void gemm_wmma_f32(const float* __restrict__ A,
                              const float* __restrict__ B,
                              float* __restrict__ C,
                              float scale) {
    const int lane = threadIdx.x & 31;
    const int half = lane >> 4;
    const int l    = lane & 15;
    const int n0   = blockIdx.x * 16;
    const int m0   = blockIdx.y * 16;

    v8f acc = {};
    for (int k0 = 0; k0 < HDIM; k0 += 4) {
        const int ka = k0 + 2 * half;
        v2f a, b;
        a.x = A[(m0 + l) * HDIM + ka];
        a.y = A[(m0 + l) * HDIM + ka + 1];
        if (TRANSB) {
            // B[k][n] = w[n*H + k] -> contiguous float pair per lane
            b.x = B[(n0 + l) * HDIM + ka];
            b.y = B[(n0 + l) * HDIM + ka + 1];
        } else {
            b.x = B[(ka)     * HDIM + (n0 + l)];
            b.y = B[(ka + 1) * HDIM + (n0 + l)];
        }
        // (neg_a, A, neg_b, B, c_mod, C, reuse_a, reuse_b)
        acc = __builtin_amdgcn_wmma_f32_16x16x4_f32(
            false, a, false, b, (short)0, acc, false, false);
    }

#pragma unroll
    for (int g = 0; g < 8; ++g) {
        const int m = m0 + g + 8 * half;
        C[m * HDIM + (n0 + l)] = acc[g] * scale;
    }
}

// ---------------------------------------------------------------------------
// K2: scores[b][j] = r[b] . x2[b][j]   (1/sqrt(H) already folded into r)
//     One thread per key row; r[b] staged in LDS (broadcast reads).
//     This is the bandwidth-bound pass over the 268 MB of input_2.
// ---------------------------------------------------------------------------
__global__ void scores_kernel(const float* __restrict__ x2,   // (B, LK, H)
                              const float* __restrict__ r,    // (B, H)
                              const float* __restrict__ m1,   // (B, 1)
                              const float* __restrict__ m2,   // (B, LK)
                              float* __restrict__ scores) {   // (B, LK)
    const int b = blockIdx.y;
    const int j = blockIdx.x * blockDim.x + threadIdx.x;

    __shared__ float rs[HDIM];
    if (threadIdx.x < HDIM) rs[threadIdx.x] = r[b * HDIM + threadIdx.x];
    __syncthreads();

    const float4* __restrict__ row =
        (const float4*)(x2 + ((size_t)b * LK + j) * HDIM);
    const float4* __restrict__ wrow = (const float4*)rs;

    float acc = 0.0f;
#pragma unroll
    for (int d = 0; d < HDIM / 4; ++d) {
        const float4 v = row[d];
        const float4 w = wrow[d];
        acc = fmaf(v.x, w.x, acc);
        acc = fmaf(v.y, w.y, acc);
        acc = fmaf(v.z, w.z, acc);
        acc = fmaf(v.w, w.w, acc);
    }
    const float m = m1[b] * m2[(size_t)b * LK + j];
    scores[(size_t)b * LK + j] = (m > 0.0f) ? acc : NEG_INF;
}

// ---------------------------------------------------------------------------
// K3: per-batch softmax over 8192 scores (in-place in the attn region),
//     apply mask, write attn, scatter-add into dist via global fp32 atomics.
//     256 threads/block, 32 keys/thread (kept in registers across barriers).
// ---------------------------------------------------------------------------
__global__ void softmax_scatter_kernel(const float* __restrict__ m1,
                                       const float* __restrict__ m2,
                                       const int* __restrict__ word_inds,
                                       float* __restrict__ dist,   // (B, VOCAB)
                                       float* __restrict__ attn) { // (B, LK), holds scores on entry
    const int b   = blockIdx.x;
    const int tid = threadIdx.x;            // 256 threads
    constexpr int KPT = LK / 256;           // 32 keys per thread

    float loc[KPT];
    float m = -3.4e38f;
#pragma unroll
    for (int i = 0; i < KPT; ++i) {
        loc[i] = attn[(size_t)b * LK + tid + 256 * i];
        m = fmaxf(m, loc[i]);
    }

    __shared__ float red[256];
    red[tid] = m;
    __syncthreads();
    for (int s = 128; s > 0; s >>= 1) {
        if (tid < s) red[tid] = fmaxf(red[tid], red[tid + s]);
        __syncthreads();
    }
    const float bmax = red[0];
    __syncthreads();

    float sum = 0.0f;
#pragma unroll
    for (int i = 0; i < KPT; ++i) {
        loc[i] = expf(loc[i] - bmax);   // exp(NEG_INF - max) underflows to 0
        sum += loc[i];
    }
    red[tid] = sum;
    __syncthreads();
    for (int s = 128; s > 0; s >>= 1) {
        if (tid < s) red[tid] += red[tid + s];
        __syncthreads();
    }
    const float inv = 1.0f / red[0];
    const float mb1 = m1[b];

#pragma unroll
    for (int i = 0; i < KPT; ++i) {
        const int j = tid + 256 * i;
        const float mask = mb1 * m2[(size_t)b * LK + j];
        const float a = loc[i] * inv * mask;
        attn[(size_t)b * LK + j] = a;
        const int idx = word_inds[(size_t)b * LK + j] - 1;
        atomicAdd(&dist[(size_t)b * VOCAB + idx], a);
    }
}

// ---------------------------------------------------------------------------
// Launch: inputs in setup_inputs() order:
//  0:input_1 1:input_2 2:mask_1 3:mask_2 4:w_q 5:w_k 6:w_v(unused)
//  7:word_inds 8:vocab_size
// d_out = [ dist (64*50257 f32) | attn (64*8192 f32) ]
// ---------------------------------------------------------------------------
extern "C" void kernel_launch(void* const* d_in, const int* in_sizes, int n_in,
                              void* d_out, int out_size, void* d_ws, size_t ws_size,
                              hipStream_t stream) {
    const float* x1    = (const float*)d_in[0];
    const float* x2    = (const float*)d_in[1];
    const float* m1    = (const float*)d_in[2];
    const float* m2    = (const float*)d_in[3];
    const float* wq    = (const float*)d_in[4];
    const float* wk    = (const float*)d_in[5];
    const int*   winds = (const int*)d_in[7];

    float* out  = (float*)d_out;
    float* dist = out;                                   // (B, VOCAB)
    float* attn = out + (size_t)BATCH * VOCAB;           // (B, LK) — also scores scratch

    float* r = (float*)d_ws;                             // (B, H)
    float* Q = r + BATCH * HDIM;                         // (B, H)

    // K0: zero dist region
    const size_t ndist = (size_t)BATCH * VOCAB;
    zero_kernel<<<(unsigned)((ndist + 1023) / 1024), 1024, 0, stream>>>(dist, ndist);

    // K1a: Q = input_1 @ w_q          (WMMA f32 16x16x4)
    gemm_wmma_f32<false><<<dim3(HDIM / 16, BATCH / 16), 32, 0, stream>>>(x1, wq, Q, 1.0f);

    // K1b: r = (Q @ w_k^T) / sqrt(H)  (WMMA f32 16x16x4)
    gemm_wmma_f32<true><<<dim3(HDIM / 16, BATCH / 16), 32, 0, stream>>>(
        Q, wk, r, 0.08838834764831845f);

    // K2: masked scores -> attn region (bandwidth-bound pass over input_2)
    scores_kernel<<<dim3(LK / 256, BATCH), 256, 0, stream>>>(x2, r, m1, m2, attn);

    // K3: softmax + mask + scatter-add into dist
    softmax_scatter_kernel<<<BATCH, 256, 0, stream>>>(m1, m2, winds, dist, attn);
}